// SparseSpatioConvLayer_15504831938949
// MI455X (gfx1250) — compile-verified
//
#include <hip/hip_runtime.h>

typedef float v2f __attribute__((ext_vector_type(2)));
typedef float v8f __attribute__((ext_vector_type(8)));

#define HH 4
#define BB 4
#define CC 32
#define VV 1000
#define LL 12
#define EE 20000
#define NN (VV * LL)   /* 12000 columns per (h,b) */
#define HBN (HH * BB)  /* 16 */
#define ALPHA_F 0.2f
#define NEG_CAP_F -9000000000000000.0f

// ---------------------------------------------------------------------------
// Kernel 1: h_t[hb][n][d] = sum_c W[h][c][d] * x[b][c][n]
// Dense f32 GEMM via V_WMMA_F32_16X16X4_F32. One 16x16 output tile per wave,
// K=32 covered by 8 WMMA steps of K=4. Output stored d-contiguous (transposed)
// so the edge-scatter stage reads 32 contiguous floats per gather.
// ---------------------------------------------------------------------------
__global__ __launch_bounds__(256) void gemm_h_wmma(const float* __restrict__ x,
                                                   const float* __restrict__ W,
                                                   float* __restrict__ h_t) {
  const int wave = (blockIdx.x * blockDim.x + threadIdx.x) >> 5;
  const int lane = threadIdx.x & 31;
  const int n_tiles = NN / 16;        // 750
  const int n_tile = wave % n_tiles;
  const int rest   = wave / n_tiles;  // [0,32)
  const int m_tile = rest & 1;        // d-half: 0 or 1
  const int hb     = rest >> 1;       // [0,16)
  if (hb >= HBN) return;              // wave-uniform (EXEC stays all-1s for WMMA)
  const int h = hb >> 2;
  const int b = hb & 3;

  const int half = lane >> 4;   // selects K pair {0,1} vs {2,3} within a k-step
  const int row  = lane & 15;   // M row for A fragment / N col for B fragment
  const float* __restrict__ Wh = W + h * (CC * CC);
  const float* __restrict__ xb = x + (size_t)b * CC * NN;
  const int n0 = n_tile * 16;
  const int m0 = m_tile * 16;

  v8f acc = {};
#pragma unroll
  for (int kk = 0; kk < CC; kk += 4) {
    const int k0 = kk + 2 * half;
    v2f af, bf;
    // A[m=d, k=c] = W[h][c][d]
    af.x = Wh[(k0 + 0) * CC + (m0 + row)];
    af.y = Wh[(k0 + 1) * CC + (m0 + row)];
    // B[k=c, n] = x[b][c][n]
    bf.x = xb[(k0 + 0) * NN + (n0 + row)];
    bf.y = xb[(k0 + 1) * NN + (n0 + row)];
    acc = __builtin_amdgcn_wmma_f32_16x16x4_f32(false, af, false, bf,
                                                (short)0, acc, false, false);
  }

  // D layout: VGPR r holds (M = r + 8*half, N = row)
  float* __restrict__ outp = h_t + (size_t)hb * NN * CC;
#pragma unroll
  for (int r = 0; r < 8; ++r) {
    const int d = m0 + r + 8 * half;
    const int n = n0 + row;
    outp[(size_t)n * CC + d] = acc[r];
  }
}

// ---------------------------------------------------------------------------
// Kernel 2: per-(hb,n) attention projections
//   s_src[hb,n] = sum_d a_src[h,d] * h_t[hb,n,d]; s_dst likewise.
// ---------------------------------------------------------------------------
__global__ void attn_scores(const float* __restrict__ h_t,
                            const float* __restrict__ a,
                            float* __restrict__ s_src,
                            float* __restrict__ s_dst) {
  const int idx = blockIdx.x * blockDim.x + threadIdx.x;
  if (idx >= HBN * NN) return;
  const int hb = idx / NN;
  const int h = hb >> 2;
  const float* __restrict__ hp = h_t + (size_t)idx * CC;
  const float* __restrict__ asrc = a + h * (2 * CC);
  const float* __restrict__ adst = asrc + CC;
  float ss = 0.f, sd = 0.f;
#pragma unroll
  for (int d = 0; d < CC; ++d) {
    const float v = hp[d];
    ss += v * asrc[d];
    sd += v * adst[d];
  }
  s_src[idx] = ss;
  s_dst[idx] = sd;
}

// ---------------------------------------------------------------------------
// Kernel 3: fused edge attention + scatter.
// One thread per (hb, e, l): compute edge_e, atomically accumulate rowsum and
// the 32-wide weighted message into num_t (both d-contiguous, L2-resident).
// ---------------------------------------------------------------------------
__global__ void edge_scatter(const float* __restrict__ s_src,
                             const float* __restrict__ s_dst,
                             const float* __restrict__ h_t,
                             const long long* __restrict__ e0,
                             const long long* __restrict__ e1,
                             const float* __restrict__ edge_val,
                             float* __restrict__ rs,
                             float* __restrict__ num_t) {
  const int idx = blockIdx.x * blockDim.x + threadIdx.x;
  if (idx >= HBN * EE * LL) return;
  const int l = idx % LL;
  const int e = (idx / LL) % EE;
  const int hb = idx / (LL * EE);
  const int v0 = (int)e0[e];
  const int v1 = (int)e1[e];
  const int i0 = hb * NN + v0 * LL + l;  // destination (segment) slot
  const int i1 = hb * NN + v1 * LL + l;  // source gather slot

  const float s = s_src[i0] + s_dst[i1];
  const float lr = (s >= 0.f) ? s : ALPHA_F * s;           // leaky_relu
  const float lv = fmaxf(__logf(edge_val[e]), NEG_CAP_F);  // log clamp
  const float ee = __expf(-lr) * lv;

  atomicAdd(&rs[i0], ee);

  const float* __restrict__ hsrc = h_t + (size_t)i1 * CC;
  float* __restrict__ nd = num_t + (size_t)i0 * CC;
#pragma unroll 8
  for (int d = 0; d < CC; ++d) {
    atomicAdd(&nd[d], ee * hsrc[d]);
  }
}

// ---------------------------------------------------------------------------
// Kernel 4: h_prime = num/rowsum, mean over heads, ELU, write (B,C,V,L).
// ---------------------------------------------------------------------------
__global__ void finalize_out(const float* __restrict__ num_t,
                             const float* __restrict__ rs,
                             float* __restrict__ out) {
  const int idx = blockIdx.x * blockDim.x + threadIdx.x;
  if (idx >= BB * CC * VV * LL) return;
  const int l = idx % LL;
  const int v = (idx / LL) % VV;
  const int c = (idx / (LL * VV)) % CC;
  const int b = idx / (LL * VV * CC);
  float acc = 0.f;
#pragma unroll
  for (int h = 0; h < HH; ++h) {
    const int hb = h * BB + b;
    const int n = hb * NN + v * LL + l;
    acc += num_t[(size_t)n * CC + c] / rs[n];
  }
  acc *= (1.0f / HH);
  out[idx] = (acc > 0.f) ? acc : (__expf(acc) - 1.f);  // ELU
}

// ---------------------------------------------------------------------------
extern "C" void kernel_launch(void* const* d_in, const int* in_sizes, int n_in,
                              void* d_out, int out_size, void* d_ws, size_t ws_size,
                              hipStream_t stream) {
  const float* x = (const float*)d_in[0];        // (B,C,V,L)
  const float* W = (const float*)d_in[1];        // (H,C,C)
  const float* a = (const float*)d_in[2];        // (H,2C,1)
  const long long* ei = (const long long*)d_in[3];  // (2,E) int64
  const float* ev = (const float*)d_in[4];       // (E,)
  float* out = (float*)d_out;

  float* ws = (float*)d_ws;
  float* h_t   = ws;                                  // 16*12000*32 = 6.144M f32
  float* num_t = h_t + (size_t)HBN * NN * CC;         // 6.144M f32
  float* s_src = num_t + (size_t)HBN * NN * CC;       // 192K f32
  float* s_dst = s_src + HBN * NN;                    // 192K f32
  float* rs    = s_dst + HBN * NN;                    // 192K f32
  // total: ~51.5 MB of workspace

  hipMemsetAsync(num_t, 0, (size_t)HBN * NN * CC * sizeof(float), stream);
  hipMemsetAsync(rs, 0, (size_t)HBN * NN * sizeof(float), stream);

  // 24000 output tiles (16 hb * 2 m-tiles * 750 n-tiles), 8 waves per block
  gemm_h_wmma<<<3000, 256, 0, stream>>>(x, W, h_t);

  attn_scores<<<(HBN * NN + 255) / 256, 256, 0, stream>>>(h_t, a, s_src, s_dst);

  edge_scatter<<<(HBN * EE * LL + 255) / 256, 256, 0, stream>>>(
      s_src, s_dst, h_t, ei, ei + EE, ev, rs, num_t);

  finalize_out<<<(BB * CC * VV * LL + 255) / 256, 256, 0, stream>>>(num_t, rs, out);
}